// MarginLoss_71717363908615
// MI455X (gfx1250) — compile-verified
//
#include <hip/hip_runtime.h>

#define BN 4096
#define DD 512
#define NTILES (BN / 16)      // 256 column tiles
#define NPAIRS (NTILES / 2)   // 128 column-tile pairs (2x register blocking)
#define LDS_STRIDE 516        // 512 + 4 pad: 16 row-reads hit 16 distinct banks

typedef __attribute__((ext_vector_type(2))) float v2f;
typedef __attribute__((ext_vector_type(8))) float v8f;

__device__ __forceinline__ float neg_inf() { return -__builtin_inff(); }

// ---------------- zero the accumulators (ws is not re-poisoned between replays) --
__global__ void init_kernel(float* accum) {
    if (threadIdx.x < 2) accum[threadIdx.x] = 0.0f;
}

// ---------------- row squared norms (== Gram diagonal) ---------------------------
__global__ void rowsq_kernel(const float* __restrict__ x, float* __restrict__ sq) {
    const int row  = blockIdx.x;
    const int lane = threadIdx.x;          // blockDim = 32 (one wave)
    const float* xr = x + (size_t)row * DD;
    float s = 0.0f;
    for (int t = lane; t < DD; t += 32) { float v = xr[t]; s += v * v; }
    #pragma unroll
    for (int off = 16; off > 0; off >>= 1) s += __shfl_xor(s, off, 32);
    if (lane == 0) sq[row] = s;
}

// ---------------- positive sampling: argmax_j (same-label & j!=i) gumbel_pos -----
__global__ void posidx_kernel(const float* __restrict__ gpos,
                              const int*   __restrict__ labels,
                              int*         __restrict__ pos_idx) {
    __shared__ float ss[256];
    __shared__ int   si[256];
    const int i   = blockIdx.x;
    const int tid = threadIdx.x;
    const int li  = labels[i];
    const float* grow = gpos + (size_t)i * BN;
    float best = neg_inf();
    int   bi   = 0;                         // jnp.argmax of all -inf -> 0
    for (int j = tid; j < BN; j += 256) {
        bool ok = (labels[j] == li) && (j != i);
        float s = ok ? grow[j] : neg_inf();
        if (s > best) { best = s; bi = j; }
    }
    ss[tid] = best; si[tid] = bi;
    __syncthreads();
    for (int off = 128; off > 0; off >>= 1) {
        if (tid < off && ss[tid + off] > ss[tid]) {
            ss[tid] = ss[tid + off]; si[tid] = si[tid + off];
        }
        __syncthreads();
    }
    if (tid == 0) pos_idx[i] = si[0];
}

// ---------------- main: fused f32 WMMA Gram + distance-weighted neg sampling -----
// epilogue for one 16x16 accumulator tile: d2 -> dist -> log q -> mask -> +gumbel
__device__ __forceinline__ void tile_epilogue(
    const v8f& acc, int j, int rowbase, int myrow16,
    const float* __restrict__ sqi_s, const int* __restrict__ lbl_s,
    const float* __restrict__ sq, const int* __restrict__ labels,
    const float* __restrict__ gneg, float best[8], int bidx[8]) {
    const float sqj = sq[j];
    const int   lj  = labels[j];
    #pragma unroll
    for (int v = 0; v < 8; ++v) {
        const int rloc = v + myrow16;
        const int row  = rowbase + rloc;
        float d2 = sqi_s[rloc] + sqj - 2.0f * acc[v];
        d2 = fmaxf(d2, 1e-4f);                      // clip(0) + clip(DIST_EPS)
        float dist = fmaxf(__builtin_sqrtf(d2), 0.5f);   // clamp(min=cutoff)
        float lq = (2.0f - (float)DD) * logf(dist)
                 - ((float)(DD - 3) * 0.5f) * logf(1.0f - 0.25f * dist * dist);
        float score = (lbl_s[rloc] == lj) ? neg_inf() : lq;
        score += gneg[(size_t)row * BN + j];
        if (score > best[v]) { best[v] = score; bidx[v] = j; }
    }
}

__global__ void __launch_bounds__(256)
negidx_kernel(const float* __restrict__ x,
              const float* __restrict__ gneg,
              const int*   __restrict__ labels,
              const float* __restrict__ sq,
              int*         __restrict__ neg_idx) {
    __shared__ float Atile[16 * LDS_STRIDE];   // 16 rows x 512 f32 (padded)
    __shared__ float sqi_s[16];
    __shared__ int   lbl_s[16];
    __shared__ float red_s[8][16];
    __shared__ int   red_i[8][16];

    const int rowbase = blockIdx.x * 16;
    const int tid  = threadIdx.x;
    const int wave = tid >> 5;
    const int lane = tid & 31;
    const int half = lane >> 4;     // 0: lanes 0-15, 1: lanes 16-31
    const int l15  = lane & 15;

    // stage A tile (row block of X) into LDS, coalesced
    for (int t = tid; t < 16 * DD; t += 256) {
        int r = t >> 9;             // /512
        int c = t & (DD - 1);
        Atile[r * LDS_STRIDE + c] = x[(size_t)(rowbase + r) * DD + c];
    }
    if (tid < 16) {
        sqi_s[tid] = sq[rowbase + tid];
        lbl_s[tid] = labels[rowbase + tid];
    }
    __syncthreads();

    float best[8];
    int   bidx[8];
    #pragma unroll
    for (int v = 0; v < 8; ++v) { best[v] = neg_inf(); bidx[v] = 0; }

    const int myrow16 = 8 * half;                       // C/D row base for this half
    const float* arow = &Atile[l15 * LDS_STRIDE + 2 * half];  // A frag gather (ISA 16x4 layout)

    // each wave sweeps 16 column-tile PAIRS (16x32 output per sweep):
    // one A fragment from LDS feeds two WMMAs -> halves DS traffic per WMMA.
    for (int cp = wave; cp < NPAIRS; cp += 8) {
        const int colbase0 = cp * 32;
        const int j0 = colbase0 + l15;                  // lane's column in tile 0
        const int j1 = j0 + 16;                         // lane's column in tile 1
        const float* bptr0 = x + (size_t)j0 * DD + 2 * half;
        const float* bptr1 = x + (size_t)j1 * DD + 2 * half;

        v8f acc0 = {0.f, 0.f, 0.f, 0.f, 0.f, 0.f, 0.f, 0.f};
        v8f acc1 = {0.f, 0.f, 0.f, 0.f, 0.f, 0.f, 0.f, 0.f};
        #pragma unroll 4
        for (int k = 0; k < DD; k += 4) {
            v2f a;  a.x  = arow[k];      a.y  = arow[k + 1];   // LDS, shared by both tiles
            v2f b0; b0.x = bptr0[k];     b0.y = bptr0[k + 1];
            v2f b1; b1.x = bptr1[k];     b1.y = bptr1[k + 1];
            acc0 = __builtin_amdgcn_wmma_f32_16x16x4_f32(
                false, a, false, b0, (short)0, acc0, false, false);
            acc1 = __builtin_amdgcn_wmma_f32_16x16x4_f32(
                false, a, false, b1, (short)0, acc1, false, false);
        }

        tile_epilogue(acc0, j0, rowbase, myrow16, sqi_s, lbl_s, sq, labels, gneg, best, bidx);
        tile_epilogue(acc1, j1, rowbase, myrow16, sqi_s, lbl_s, sq, labels, gneg, best, bidx);
    }

    // argmax butterfly within each 16-lane half (wave32; masks 1..8 stay in-half)
    #pragma unroll
    for (int off = 1; off < 16; off <<= 1) {
        #pragma unroll
        for (int v = 0; v < 8; ++v) {
            float os = __shfl_xor(best[v], off, 32);
            int   oi = __shfl_xor(bidx[v], off, 32);
            if (os > best[v]) { best[v] = os; bidx[v] = oi; }
        }
    }
    if (l15 == 0) {
        #pragma unroll
        for (int v = 0; v < 8; ++v) {
            red_s[wave][v + myrow16] = best[v];
            red_i[wave][v + myrow16] = bidx[v];
        }
    }
    __syncthreads();
    if (tid < 16) {
        float bs = red_s[0][tid]; int bi = red_i[0][tid];
        #pragma unroll
        for (int w = 1; w < 8; ++w)
            if (red_s[w][tid] > bs) { bs = red_s[w][tid]; bi = red_i[w][tid]; }
        neg_idx[rowbase + tid] = bi;
    }
}

// ---------------- triplet margin loss + global accumulation ----------------------
__global__ void loss_kernel(const float* __restrict__ x,
                            const float* __restrict__ beta,
                            const int*   __restrict__ labels,
                            const int*   __restrict__ pos_idx,
                            const int*   __restrict__ neg_idx,
                            float* accum) {
    const int wave = threadIdx.x >> 5;
    const int lane = threadIdx.x & 31;
    const int i = blockIdx.x * 8 + wave;            // one wave per row
    const float* xi = x + (size_t)i * DD;
    const float* xp = x + (size_t)pos_idx[i] * DD;
    const float* xn = x + (size_t)neg_idx[i] * DD;
    float sp = 0.f, sn = 0.f;
    for (int t = lane; t < DD; t += 32) {
        float dp = xi[t] - xp[t]; sp += dp * dp;
        float dn = xi[t] - xn[t]; sn += dn * dn;
    }
    #pragma unroll
    for (int off = 16; off > 0; off >>= 1) {
        sp += __shfl_xor(sp, off, 32);
        sn += __shfl_xor(sn, off, 32);
    }
    if (lane == 0) {
        float dap = __builtin_sqrtf(sp + 1e-8f);
        float dan = __builtin_sqrtf(sn + 1e-8f);
        float b   = beta[labels[i]];
        float pl  = fmaxf(dap - b + 0.2f, 0.f);
        float nl  = fmaxf(b - dan + 0.2f, 0.f);
        float cnt = (pl > 0.f ? 1.f : 0.f) + (nl > 0.f ? 1.f : 0.f);
        atomicAdd(&accum[0], pl + nl);
        atomicAdd(&accum[1], cnt);
    }
}

__global__ void finalize_kernel(const float* accum, float* out) {
    float total = accum[0], cnt = accum[1];
    out[0] = (cnt == 0.f) ? total : total / cnt;
}

// ---------------------------------------------------------------------------------
extern "C" void kernel_launch(void* const* d_in, const int* in_sizes, int n_in,
                              void* d_out, int out_size, void* d_ws, size_t ws_size,
                              hipStream_t stream) {
    const float* batch  = (const float*)d_in[0];  // [4096,512]
    const float* beta   = (const float*)d_in[1];  // [100]
    const float* gpos   = (const float*)d_in[2];  // [4096,4096]
    const float* gneg   = (const float*)d_in[3];  // [4096,4096]
    const int*   labels = (const int*)  d_in[4];  // [4096]
    float* out = (float*)d_out;

    char* ws = (char*)d_ws;
    float* sq      = (float*)(ws);                 // 4096 f32
    int*   pos_idx = (int*)  (ws + 4096 * 4);      // 4096 i32
    int*   neg_idx = (int*)  (ws + 4096 * 8);      // 4096 i32
    float* accum   = (float*)(ws + 4096 * 12);     // 2 f32

    init_kernel    <<<1,       32,  0, stream>>>(accum);
    rowsq_kernel   <<<BN,      32,  0, stream>>>(batch, sq);
    posidx_kernel  <<<BN,      256, 0, stream>>>(gpos, labels, pos_idx);
    negidx_kernel  <<<BN / 16, 256, 0, stream>>>(batch, gneg, labels, sq, neg_idx);
    loss_kernel    <<<BN / 8,  256, 0, stream>>>(batch, beta, labels, pos_idx, neg_idx, accum);
    finalize_kernel<<<1,       1,   0, stream>>>(accum, out);
}